// PositionSetLoss_41154376630568
// MI455X (gfx1250) — compile-verified
//
#include <hip/hip_runtime.h>
#include <hip/hip_bf16.h>

typedef __attribute__((ext_vector_type(2))) float v2f;
typedef __attribute__((ext_vector_type(8))) float v8f;

#define N1 16384
#define N2 16384
#define NCHUNK 8
#define CHUNK (N2 / NCHUNK)            // 2048 j per chunk
#define TILES_PER_CHUNK (CHUNK / 16)   // 128 WMMA tiles per chunk
#define WAVES_PER_BLOCK 8

// Workspace layout:
//   [0        , N1*4     ) : wsmin  (uint per i, bit pattern of min dist2)
//   [64K      , 64K+256K ) : aprep  (float2[2][N2], A-operand in WMMA lane layout)
#define WSMIN_BYTES  (N1 * 4)
#define APREP_OFFSET 65536
#define WS_NEEDED    (APREP_OFFSET + 2 * N2 * 8)

__global__ void nn_init_kernel(unsigned int* __restrict__ wsmin) {
    int i = blockIdx.x * blockDim.x + threadIdx.x;
    if (i < N1) wsmin[i] = 0x7F800000u;  // +inf
}

// Materialize the A matrix operand in exact WMMA lane layout, once per j:
//   a0[j] = (-2*x2, -2*y2)   -> K=0,1 (consumed by lanes 0-15)
//   a1[j] = (||p2||^2, 0)    -> K=2,3 (consumed by lanes 16-31)
__global__ void nn_prep_kernel(const float* __restrict__ pos2, float2* __restrict__ aprep) {
    int j = blockIdx.x * blockDim.x + threadIdx.x;
    if (j < N2) {
        float2 p2 = ((const float2*)pos2)[j];
        float2 lo, hi;
        lo.x = -2.0f * p2.x;
        lo.y = -2.0f * p2.y;
        hi.x = p2.x * p2.x + p2.y * p2.y;
        hi.y = 0.0f;
        aprep[j]      = lo;
        aprep[N2 + j] = hi;
    }
}

// One wave handles one (i-tile of 16 columns, j-chunk of 2048 rows).
// dist2[i,j] = ||p1||^2 + ( ||p2||^2 - 2*p1.p2 ); the parenthesized part comes
// straight out of V_WMMA_F32_16X16X4_F32 with:
//   A (16x4, M=j rows):  [ -2*x2, -2*y2, ||p2||^2, 0 ]   (preloaded from aprep)
//   B (4x16, N=i cols):  [   x1 ,   y1 ,    1    , 0 ]^T
// D layout: lane l, vgpr r -> (M = r + 8*(l>=16), N = l&15), so min over j is
// in-lane mins per tile + one cross-half shuffle at the end.
__global__ __launch_bounds__(256) void nn_wmma_kernel(const float2* __restrict__ aprep,
                                                      const float* __restrict__ pos1,
                                                      unsigned int* __restrict__ wsmin) {
    const int lane = threadIdx.x & 31;
    const int wid  = threadIdx.x >> 5;
    const int gw   = blockIdx.x * WAVES_PER_BLOCK + wid;   // 0 .. 8191
    const int itile = gw >> 3;                             // / NCHUNK
    const int chunk = gw & (NCHUNK - 1);
    const int n  = lane & 15;
    const bool hi = (lane >= 16);                          // lanes 16-31 carry K=2,3

    // --- B operand (constant over the whole j loop) ---
    const float2 p1 = ((const float2*)pos1)[itile * 16 + n];
    const float n1v = p1.x * p1.x + p1.y * p1.y;           // ||p1||^2, added after min
    v2f B;
    B.x = hi ? 1.0f : p1.x;   // K=0 (lanes 0-15) / K=2 (lanes 16-31)
    B.y = hi ? 0.0f : p1.y;   // K=1              / K=3

    // Per-lane A source: lanes 0-15 read aprep[0..N2), lanes 16-31 read aprep[N2..2*N2)
    const v2f* __restrict__ abase = (const v2f*)aprep + (hi ? N2 : 0);

    v8f acc;
#pragma unroll
    for (int r = 0; r < 8; ++r) acc[r] = __builtin_inff();

    const int jbase0 = chunk * CHUNK + n;
#pragma unroll 4
    for (int t = 0; t < TILES_PER_CHUNK; ++t) {
        v2f A = abase[jbase0 + t * 16];                    // one b64 load, zero VALU prep
        v8f c = {};
        // D = A x B + 0  ->  D[m,n] = ||p2_m||^2 - 2 * p1_n . p2_m
        v8f d = __builtin_amdgcn_wmma_f32_16x16x4_f32(
            /*neg_a=*/false, A, /*neg_b=*/false, B,
            /*c_mod=*/(short)0, c, /*reuse_a=*/false, /*reuse_b=*/false);
#pragma unroll
        for (int r = 0; r < 8; ++r) acc[r] = fminf(acc[r], d[r]);
    }

    // Reduce over the 8 M-rows held in this lane, then across the two halves.
    float m = acc[0];
#pragma unroll
    for (int r = 1; r < 8; ++r) m = fminf(m, acc[r]);
    m = fminf(m, __shfl_xor(m, 16, 32));      // combine M=0..7 with M=8..15

    const float d2 = fmaxf(m + n1v, 0.0f);    // clamp tiny negative cancellation
    if (lane < 16) {
        atomicMin(&wsmin[itile * 16 + n], __float_as_uint(d2));
    }
}

// Fallback (ws too small for aprep): build A in-loop.
__global__ __launch_bounds__(256) void nn_wmma_kernel_nopre(const float* __restrict__ pos1,
                                                            const float* __restrict__ pos2,
                                                            unsigned int* __restrict__ wsmin) {
    const int lane = threadIdx.x & 31;
    const int wid  = threadIdx.x >> 5;
    const int gw   = blockIdx.x * WAVES_PER_BLOCK + wid;
    const int itile = gw >> 3;
    const int chunk = gw & (NCHUNK - 1);
    const int n  = lane & 15;
    const bool hi = (lane >= 16);

    const float2 p1 = ((const float2*)pos1)[itile * 16 + n];
    const float n1v = p1.x * p1.x + p1.y * p1.y;
    v2f B;
    B.x = hi ? 1.0f : p1.x;
    B.y = hi ? 0.0f : p1.y;

    v8f acc;
#pragma unroll
    for (int r = 0; r < 8; ++r) acc[r] = __builtin_inff();

    const int jbase0 = chunk * CHUNK;
    for (int t = 0; t < TILES_PER_CHUNK; ++t) {
        const float2 p2 = ((const float2*)pos2)[jbase0 + t * 16 + n];
        const float n2v = p2.x * p2.x + p2.y * p2.y;
        v2f A;
        A.x = hi ? n2v  : (-2.0f * p2.x);
        A.y = hi ? 0.0f : (-2.0f * p2.y);
        v8f c = {};
        v8f d = __builtin_amdgcn_wmma_f32_16x16x4_f32(
            false, A, false, B, (short)0, c, false, false);
#pragma unroll
        for (int r = 0; r < 8; ++r) acc[r] = fminf(acc[r], d[r]);
    }

    float m = acc[0];
#pragma unroll
    for (int r = 1; r < 8; ++r) m = fminf(m, acc[r]);
    m = fminf(m, __shfl_xor(m, 16, 32));
    const float d2 = fmaxf(m + n1v, 0.0f);
    if (lane < 16) {
        atomicMin(&wsmin[itile * 16 + n], __float_as_uint(d2));
    }
}

__global__ void nn_reduce_kernel(const unsigned int* __restrict__ wsmin,
                                 float* __restrict__ out) {
    __shared__ float sm[256];
    float s = 0.0f;
    for (int i = threadIdx.x; i < N1; i += 256)
        s += sqrtf(__uint_as_float(wsmin[i]));   // sqrt(min d2) == min sqrt(d2)
    sm[threadIdx.x] = s;
    __syncthreads();
    for (int off = 128; off > 0; off >>= 1) {
        if (threadIdx.x < off) sm[threadIdx.x] += sm[threadIdx.x + off];
        __syncthreads();
    }
    if (threadIdx.x == 0) out[0] = sm[0] * (1.0f / (float)N1);
}

extern "C" void kernel_launch(void* const* d_in, const int* in_sizes, int n_in,
                              void* d_out, int out_size, void* d_ws, size_t ws_size,
                              hipStream_t stream) {
    const float* pos1 = (const float*)d_in[0];   // [16384, 2] f32
    const float* pos2 = (const float*)d_in[1];   // [16384, 2] f32
    float* out = (float*)d_out;                  // scalar f32
    unsigned int* wsmin = (unsigned int*)d_ws;

    nn_init_kernel<<<(N1 + 255) / 256, 256, 0, stream>>>(wsmin);

    const int total_waves = (N1 / 16) * NCHUNK;  // 8192 waves
    if (ws_size >= (size_t)WS_NEEDED) {
        float2* aprep = (float2*)((char*)d_ws + APREP_OFFSET);
        nn_prep_kernel<<<(N2 + 255) / 256, 256, 0, stream>>>(pos2, aprep);
        nn_wmma_kernel<<<total_waves / WAVES_PER_BLOCK, 32 * WAVES_PER_BLOCK, 0, stream>>>(
            aprep, pos1, wsmin);
    } else {
        nn_wmma_kernel_nopre<<<total_waves / WAVES_PER_BLOCK, 32 * WAVES_PER_BLOCK, 0, stream>>>(
            pos1, pos2, wsmin);
    }

    nn_reduce_kernel<<<1, 256, 0, stream>>>(wsmin, out);
}